// PointNetFeaturePropagation_10874857193496
// MI455X (gfx1250) — compile-verified
//
#include <hip/hip_runtime.h>
#include <hip/hip_bf16.h>

typedef __attribute__((ext_vector_type(2))) float v2f;
typedef __attribute__((ext_vector_type(8))) float v8f;

#define B_    16
#define N_    4096
#define G_    1024
#define D1_   128
#define D2_   256
#define CIN_  384
#define CH_   256
#define COUT_ 256
#define P_    (B_ * N_)        // 65536 points total
#define BN_EPS 1e-5f

// ---------------------------------------------------------------------------
// Kernel 1: 3-NN inverse-distance interpolation + concat -> X [P, 384]
// One thread per point; coord2 for the batch staged in LDS (12 KB).
// ---------------------------------------------------------------------------
__global__ void interp_concat_kernel(const float* __restrict__ f1,   // [B,128,N]
                                     const float* __restrict__ c1,   // [B,3,N]
                                     const float* __restrict__ f2,   // [B,256,G]
                                     const float* __restrict__ c2,   // [B,3,G]
                                     float* __restrict__ X)          // [P,384]
{
    __shared__ float s2x[G_], s2y[G_], s2z[G_];
    const int tid = threadIdx.x;
    const int b  = blockIdx.x >> 4;          // 16 chunks of 256 points per batch
    const int n0 = (blockIdx.x & 15) << 8;

    const float* c2b = c2 + (size_t)b * 3 * G_;
    for (int i = tid; i < G_; i += 256) {
        s2x[i] = c2b[i];
        s2y[i] = c2b[G_ + i];
        s2z[i] = c2b[2 * G_ + i];
    }
    __syncthreads();

    const int n = n0 + tid;
    const float* c1b = c1 + (size_t)b * 3 * N_;
    const float px = c1b[n];
    const float py = c1b[N_ + n];
    const float pz = c1b[2 * N_ + n];

    float d0 = 3.4e38f, d1 = 3.4e38f, d2 = 3.4e38f;
    int   i0 = 0, i1 = 0, i2 = 0;
    for (int g = 0; g < G_; ++g) {
        const float dx = px - s2x[g];
        const float dy = py - s2y[g];
        const float dz = pz - s2z[g];
        const float d  = dx * dx + dy * dy + dz * dz;
        if (d < d0)      { d2 = d1; i2 = i1; d1 = d0; i1 = i0; d0 = d; i0 = g; }
        else if (d < d1) { d2 = d1; i2 = i1; d1 = d;  i1 = g; }
        else if (d < d2) { d2 = d;  i2 = g; }
    }
    const float r0 = 1.f / (d0 + 1e-8f);
    const float r1 = 1.f / (d1 + 1e-8f);
    const float r2 = 1.f / (d2 + 1e-8f);
    const float rs = 1.f / (r0 + r1 + r2);
    const float w0 = r0 * rs, w1 = r1 * rs, w2 = r2 * rs;

    const size_t p = (size_t)b * N_ + n;
    float* xr = X + p * CIN_;

    // copy feature1 (channel-major in, point-major out)
    const float* f1b = f1 + (size_t)b * D1_ * N_ + n;
    #pragma unroll 4
    for (int c = 0; c < D1_; ++c) xr[c] = f1b[(size_t)c * N_];

    // weighted gather of 3 neighbor feature rows (L2-resident: 16.8 MB)
    const float* f2b = f2 + (size_t)b * D2_ * G_;
    #pragma unroll 4
    for (int d = 0; d < D2_; ++d) {
        const float* col = f2b + (size_t)d * G_;
        xr[D1_ + d] = w0 * col[i0] + w1 * col[i1] + w2 * col[i2];
    }
}

// ---------------------------------------------------------------------------
// Templated fp32 WMMA GEMM:  Y[p, c] = sum_k W[c,k] * xin[p,k] + bias[c]
// KTOT = inner dim (multiple of 64); BN: apply y*scale+shift, relu while
// staging the input tile (fuses BN1+ReLU of layer 1 into layer 2's load).
// Workgroup: 8 waves -> tile of 128 channels x 64 points.
// ---------------------------------------------------------------------------
template <int KTOT, bool BN>
__global__ __launch_bounds__(256)
void gemm_wmma_kernel(const float* __restrict__ W,      // [C, KTOT] row-major
                      const float* __restrict__ bias,   // [C]
                      const float* __restrict__ xin,    // [P, KTOT] row-major
                      const float* __restrict__ scale,  // [KTOT] (BN only)
                      const float* __restrict__ shift,  // [KTOT] (BN only)
                      float* __restrict__ yout)         // [P, 256]
{
    constexpr int KC  = 64;       // K chunk staged in LDS
    constexpr int LDP = KC + 4;   // pad: stride 68 -> lanes hit distinct banks

    __shared__ float lw[128 * LDP];   // 34816 B
    __shared__ float lx[64 * LDP];    // 17408 B

    const int tid  = threadIdx.x;
    const int lane = tid & 31;
    const int wv   = tid >> 5;
    const int pb   = blockIdx.x * 64;    // point-tile base
    const int cb   = blockIdx.y * 128;   // channel-tile base
    const int mo   = wv * 16;            // this wave's channel sub-tile

    const int m  = lane & 15;            // A-row / B-col index for this lane
    const int kh = (lane >> 4) << 1;     // K pair: lanes 0-15 -> K 0,1; 16-31 -> K 2,3

    v8f acc[4] = {};                     // 4 x (16x16 f32) = this wave's 16x64 tile

    for (int kc = 0; kc < KTOT; kc += KC) {
        __syncthreads();
        // stage W chunk: 128 x 64, coalesced
        for (int i = tid; i < 128 * KC; i += 256) {
            const int r = i >> 6, c = i & 63;
            lw[r * LDP + c] = W[(size_t)(cb + r) * KTOT + kc + c];
        }
        // stage input chunk: 64 x 64, coalesced (+ fused BN1/ReLU for layer 2)
        for (int i = tid; i < 64 * KC; i += 256) {
            const int r = i >> 6, c = i & 63;
            float v = xin[(size_t)(pb + r) * KTOT + kc + c];
            if (BN) v = fmaxf(v * scale[kc + c] + shift[kc + c], 0.f);
            lx[r * LDP + c] = v;
        }
        __syncthreads();

        // hint next K chunk toward the caches (global_prefetch_b8)
        if (kc + KC < KTOT) {
            __builtin_prefetch(&xin[(size_t)(pb + (tid & 63)) * KTOT + kc + KC], 0, 0);
            __builtin_prefetch(&W[(size_t)(cb + (tid & 127)) * KTOT + kc + KC], 0, 0);
        }

        #pragma unroll
        for (int k0 = 0; k0 < KC; k0 += 4) {
            v2f a;                                    // A: 16x4 of W
            const float* ap = &lw[(mo + m) * LDP + k0 + kh];
            a.x = ap[0]; a.y = ap[1];
            #pragma unroll
            for (int nt = 0; nt < 4; ++nt) {
                v2f bf;                               // B: 4x16 of X^T
                const float* bp = &lx[(nt * 16 + m) * LDP + k0 + kh];
                bf.x = bp[0]; bf.y = bp[1];
                acc[nt] = __builtin_amdgcn_wmma_f32_16x16x4_f32(
                    /*neg_a=*/false, a, /*neg_b=*/false, bf,
                    /*c_mod=*/(short)0, acc[nt],
                    /*reuse_a=*/false, /*reuse_b=*/false);
            }
        }
    }

    // D layout (ISA 7.12.2): lane n = lane&15; VGPR r -> M = r (+8 for lanes>=16)
    const int mbase = (lane >> 4) ? 8 : 0;
    #pragma unroll
    for (int nt = 0; nt < 4; ++nt) {
        const size_t p = (size_t)pb + nt * 16 + m;
        float* yr = yout + p * CH_;
        #pragma unroll
        for (int r = 0; r < 8; ++r) {
            const int c = cb + mo + mbase + r;
            yr[c] = acc[nt][r] + bias[c];
        }
    }
}

// ---------------------------------------------------------------------------
// Per-channel sum / sum-of-squares over all P points (one block per channel).
// ---------------------------------------------------------------------------
__global__ void bn_stats_kernel(const float* __restrict__ y, // [P, 256]
                                float* __restrict__ sum,
                                float* __restrict__ sq)
{
    const int c = blockIdx.x;
    const int tid = threadIdx.x;
    float s = 0.f, q = 0.f;
    for (int p = tid; p < P_; p += 256) {
        const float v = y[(size_t)p * CH_ + c];
        s += v; q += v * v;
    }
    __shared__ float ss[256], qq[256];
    ss[tid] = s; qq[tid] = q;
    __syncthreads();
    for (int o = 128; o > 0; o >>= 1) {
        if (tid < o) { ss[tid] += ss[tid + o]; qq[tid] += qq[tid + o]; }
        __syncthreads();
    }
    if (tid == 0) { sum[c] = ss[0]; sq[c] = qq[0]; }
}

__global__ void bn_final_kernel(const float* __restrict__ sum,
                                const float* __restrict__ sq,
                                const float* __restrict__ gamma,
                                const float* __restrict__ beta,
                                float* __restrict__ scale,
                                float* __restrict__ shift)
{
    const int c = threadIdx.x;
    const float m  = sum[c] * (1.f / (float)P_);
    const float v  = sq[c] * (1.f / (float)P_) - m * m;
    const float sc = gamma[c] * rsqrtf(v + BN_EPS);
    scale[c] = sc;
    shift[c] = beta[c] - m * sc;
}

// ---------------------------------------------------------------------------
// Apply BN2 + ReLU and transpose [P,256] -> out [B, 256, N]
// ---------------------------------------------------------------------------
__global__ void finalize_out_kernel(const float* __restrict__ y2,
                                    const float* __restrict__ scale,
                                    const float* __restrict__ shift,
                                    float* __restrict__ out)
{
    const int n = (blockIdx.x << 8) + threadIdx.x;   // 16 blocks of 256 n
    const int c = blockIdx.y;
    const int b = blockIdx.z;
    float v = y2[((size_t)b * N_ + n) * CH_ + c];
    v = fmaxf(v * scale[c] + shift[c], 0.f);
    out[((size_t)b * COUT_ + c) * N_ + n] = v;
}

// ---------------------------------------------------------------------------
extern "C" void kernel_launch(void* const* d_in, const int* in_sizes, int n_in,
                              void* d_out, int out_size, void* d_ws, size_t ws_size,
                              hipStream_t stream)
{
    const float* feature1 = (const float*)d_in[0];
    const float* coord1   = (const float*)d_in[1];
    const float* feature2 = (const float*)d_in[2];
    const float* coord2   = (const float*)d_in[3];
    const float* W1 = (const float*)d_in[4];
    const float* b1 = (const float*)d_in[5];
    const float* g1 = (const float*)d_in[6];
    const float* be1 = (const float*)d_in[7];
    const float* W2 = (const float*)d_in[8];
    const float* b2 = (const float*)d_in[9];
    const float* g2 = (const float*)d_in[10];
    const float* be2 = (const float*)d_in[11];
    float* out = (float*)d_out;

    // workspace carve-up (floats)
    float* X    = (float*)d_ws;            // P * 384
    float* Y1   = X  + (size_t)P_ * CIN_;  // P * 256
    float* Y2   = Y1 + (size_t)P_ * CH_;   // P * 256
    float* st   = Y2 + (size_t)P_ * CH_;   // stats block
    float* sum1 = st;          float* sq1 = st + 256;
    float* sc1  = st + 512;    float* sh1 = st + 768;
    float* sum2 = st + 1024;   float* sq2 = st + 1280;
    float* sc2  = st + 1536;   float* sh2 = st + 1792;

    // 1) 3-NN interpolation + concat -> X [P, 384]
    interp_concat_kernel<<<dim3(B_ * (N_ / 256)), dim3(256), 0, stream>>>(
        feature1, coord1, feature2, coord2, X);

    // 2) layer-1 GEMM (fp32 WMMA): Y1 = W1 @ X + b1
    gemm_wmma_kernel<CIN_, false><<<dim3(P_ / 64, CH_ / 128), dim3(256), 0, stream>>>(
        W1, b1, X, nullptr, nullptr, Y1);

    // 3) BN1 batch stats + scale/shift
    bn_stats_kernel<<<dim3(CH_), dim3(256), 0, stream>>>(Y1, sum1, sq1);
    bn_final_kernel<<<dim3(1), dim3(256), 0, stream>>>(sum1, sq1, g1, be1, sc1, sh1);

    // 4) layer-2 GEMM with fused BN1+ReLU on the input: Y2 = W2 @ relu(bn(Y1)) + b2
    gemm_wmma_kernel<CH_, true><<<dim3(P_ / 64, COUT_ / 128), dim3(256), 0, stream>>>(
        W2, b2, Y1, sc1, sh1, Y2);

    // 5) BN2 batch stats + scale/shift
    bn_stats_kernel<<<dim3(COUT_), dim3(256), 0, stream>>>(Y2, sum2, sq2);
    bn_final_kernel<<<dim3(1), dim3(256), 0, stream>>>(sum2, sq2, g2, be2, sc2, sh2);

    // 6) apply BN2 + ReLU, transpose to [B, 256, N]
    finalize_out_kernel<<<dim3(N_ / 256, COUT_, B_), dim3(256), 0, stream>>>(
        Y2, sc2, sh2, out);
}